// GraphPruning_41171556499899
// MI455X (gfx1250) — compile-verified
//
#include <hip/hip_runtime.h>

#define NNODES 16384
#define KNN 16
#define NGRAPH 8

typedef float v2f __attribute__((ext_vector_type(2)));
typedef float v8f __attribute__((ext_vector_type(8)));

// Intra-wave LDS synchronization: single wave per block executes in lockstep,
// so waiting for outstanding DS ops is sufficient (no workgroup barrier needed).
__device__ __forceinline__ void wave_lds_fence() {
    asm volatile("s_wait_dscnt 0" ::: "memory");
}

// ---------------------------------------------------------------------------
// Graph boundary detection: S[g] = first index i with batch[i] >= g, else N.
// batch is sorted, so each S[g] is written by exactly one thread (no atomics).
// ---------------------------------------------------------------------------
__global__ void k_init_bounds(int* __restrict__ S) {
    int t = threadIdx.x;
    if (t <= NGRAPH) S[t] = NNODES;
}

__global__ void k_find_bounds(const int* __restrict__ batch, int* __restrict__ S) {
    int i = blockIdx.x * blockDim.x + threadIdx.x;
    if (i >= NNODES) return;
    int b  = batch[i];
    int bp = (i == 0) ? -1 : batch[i - 1];
    for (int g = bp + 1; g <= b; ++g) S[g] = i;
}

// ---------------------------------------------------------------------------
// kNN: one wave (32 threads) owns 32 consecutive query nodes. Candidate tiles
// of 16 are processed with two V_WMMA_F32_16X16X4_F32 Gram-tile computations
// (queries x candidates, K = {x,y,z,0}); each lane then maintains a private
// sorted top-16 (distance, index) list in registers.
// ---------------------------------------------------------------------------
__global__ void __launch_bounds__(32) k_knn(const float* __restrict__ pos,
                                            const int* __restrict__ batch,
                                            const int* __restrict__ S,
                                            int* __restrict__ nbr) {
    __shared__ float sGram[32 * 17];   // padded stride: avoid LDS bank conflicts
    __shared__ float sCs[16];          // candidate |c|^2
    __shared__ int   sCb[16];          // candidate batch id

    const int lane = threadIdx.x;      // 0..31
    const int idx  = lane & 15;
    const int half = lane >> 4;
    const int q0   = blockIdx.x * 32;
    const int q    = q0 + lane;        // query owned by this lane

    // A operands (16x4 f32 WMMA layout):
    //   lanes 0-15 : VGPR0=K0, VGPR1=K1 -> (x, y) of row M = idx
    //   lanes 16-31: VGPR0=K2, VGPR1=K3 -> (z, 0) of row M = idx
    const float* pa0 = pos + (size_t)(q0 + idx) * 3;
    const float* pa1 = pos + (size_t)(q0 + 16 + idx) * 3;
    v2f aLo, aHi;
    if (half == 0) { aLo.x = pa0[0]; aLo.y = pa0[1]; aHi.x = pa1[0]; aHi.y = pa1[1]; }
    else           { aLo.x = pa0[2]; aLo.y = 0.0f;   aHi.x = pa1[2]; aHi.y = 0.0f;   }

    const float qx = pos[(size_t)q * 3 + 0];
    const float qy = pos[(size_t)q * 3 + 1];
    const float qz = pos[(size_t)q * 3 + 2];
    const float sqq = qx * qx + qy * qy + qz * qz;
    const int   bq  = batch[q];

    // Candidate range for this wave: union of its queries' graphs (batch sorted).
    const int lo = S[batch[q0]];
    const int hi = S[batch[q0 + 31] + 1];

    float kd[KNN];
    int   ki[KNN];
    #pragma unroll
    for (int j = 0; j < KNN; ++j) { kd[j] = 3.0e38f; ki[j] = -1; }

    for (int c0 = (lo & ~15); c0 < hi; c0 += 16) {
        const int cc = c0 + idx;
        const int cl = (cc < NNODES) ? cc : (NNODES - 1);
        const float* pc = pos + (size_t)cl * 3;

        v2f b;                         // B operand: same per-lane swizzle as A
        if (half == 0) {
            float cx = pc[0], cy = pc[1], cz = pc[2];
            b.x = cx; b.y = cy;
            sCs[idx] = cx * cx + cy * cy + cz * cz;
            sCb[idx] = (cc < NNODES) ? batch[cl] : -1;
        } else {
            b.x = pc[2]; b.y = 0.0f;
        }

        v8f acc0 = {};
        v8f acc1 = {};
        acc0 = __builtin_amdgcn_wmma_f32_16x16x4_f32(false, aLo, false, b,
                                                     (short)0, acc0, false, false);
        acc1 = __builtin_amdgcn_wmma_f32_16x16x4_f32(false, aHi, false, b,
                                                     (short)0, acc1, false, false);

        // D layout: VGPR v -> row (v + 8*half), col idx
        #pragma unroll
        for (int v = 0; v < 8; ++v) {
            sGram[(v + 8 * half) * 17 + idx]      = acc0[v];
            sGram[(16 + v + 8 * half) * 17 + idx] = acc1[v];
        }
        wave_lds_fence();

        #pragma unroll 4
        for (int i = 0; i < 16; ++i) {
            const int   c  = c0 + i;
            const float d2 = sqq + sCs[i] - 2.0f * sGram[lane * 17 + i];
            const bool  ok = (c < NNODES) && (c != q) && (sCb[i] == bq);
            if (ok && d2 < kd[KNN - 1]) {
                // Branchless sorted insert (fully unrolled -> pure VGPR csel).
                #pragma unroll
                for (int j = KNN - 1; j >= 0; --j) {
                    if (kd[j] > d2) {
                        const bool fromPrev = (j > 0) && (kd[j - 1] > d2);
                        const float nd = fromPrev ? kd[j - 1] : d2;
                        const int   ni = fromPrev ? ki[j - 1] : c;
                        kd[j] = nd; ki[j] = ni;
                    }
                }
            }
        }
        wave_lds_fence();              // WAR: reads done before next tile's stores
    }

    #pragma unroll
    for (int j = 0; j < KNN; ++j) nbr[(size_t)q * KNN + j] = ki[j];
}

// ---------------------------------------------------------------------------
// Edge keep mask: edge (r,c) kept iff r in kNN(c) and c in kNN(r).
// Neighbor table is 1 MB -> fully L2 resident.
// ---------------------------------------------------------------------------
__global__ void k_edge_mask(const int* __restrict__ ei,
                            const int* __restrict__ nbr,
                            unsigned char* __restrict__ mask, int E) {
    int e = blockIdx.x * blockDim.x + threadIdx.x;
    if (e >= E) return;
    const int r = ei[e];
    const int c = ei[E + e];
    const int4* lc = (const int4*)(nbr + (size_t)c * KNN);
    const int4* lr = (const int4*)(nbr + (size_t)r * KNN);
    bool inC = false, inR = false;
    #pragma unroll
    for (int t = 0; t < 4; ++t) {
        int4 a = lc[t];
        inC = inC || (a.x == r) || (a.y == r) || (a.z == r) || (a.w == r);
        int4 b = lr[t];
        inR = inR || (b.x == c) || (b.y == c) || (b.z == c) || (b.w == c);
    }
    mask[e] = (inC && inR) ? 1 : 0;
}

// ---------------------------------------------------------------------------
// Apply mask to edge_attr (memory-bound floor pass): float4 vectorized.
// ---------------------------------------------------------------------------
__global__ void k_apply(const float* __restrict__ attr,
                        const unsigned char* __restrict__ mask,
                        float* __restrict__ out, int E) {
    int gid = blockIdx.x * blockDim.x + threadIdx.x;
    const int total = E * 16;          // E * 64 floats / 4
    if (gid >= total) return;
    const int e = gid >> 4;
    const float m = (float)mask[e];
    const float4* a4 = (const float4*)attr;
    float4* o4 = (float4*)out;
    float4 v = a4[gid];
    v.x *= m; v.y *= m; v.z *= m; v.w *= m;
    o4[gid] = v;
}

extern "C" void kernel_launch(void* const* d_in, const int* in_sizes, int n_in,
                              void* d_out, int out_size, void* d_ws, size_t ws_size,
                              hipStream_t stream) {
    const float* pos   = (const float*)d_in[0];
    const int*   batch = (const int*)d_in[1];
    const int*   ei    = (const int*)d_in[2];
    const float* attr  = (const float*)d_in[3];
    float*       out   = (float*)d_out;
    const int    E     = in_sizes[2] / 2;

    char* ws = (char*)d_ws;
    int*  S    = (int*)ws;                                    // 9 ints
    int*  nbr  = (int*)(ws + 128);                            // N*16 ints = 1 MB
    unsigned char* mask =
        (unsigned char*)(ws + 128 + (size_t)NNODES * KNN * sizeof(int));  // E bytes

    k_init_bounds<<<1, 32, 0, stream>>>(S);
    k_find_bounds<<<NNODES / 256, 256, 0, stream>>>(batch, S);
    k_knn<<<NNODES / 32, 32, 0, stream>>>(pos, batch, S, nbr);
    k_edge_mask<<<(E + 255) / 256, 256, 0, stream>>>(ei, nbr, mask, E);
    const int total = E * 16;
    k_apply<<<(total + 255) / 256, 256, 0, stream>>>(attr, mask, out, E);
}